// RA_VQVAE_Encoder_2937757630647
// MI455X (gfx1250) — compile-verified
//
#include <hip/hip_runtime.h>
#include <math.h>
#include <stdint.h>

// ---------------------------------------------------------------------------
// RA-VQVAE encoder for MI455X (gfx1250, wave32, WMMA + async global->LDS).
// N=16384 rows, D=1024 dim, K=8192 codes. fp32 throughout (memory/L2-bound
// at ~570 GFLOP; fp32 WMMA 16x16x4 is plenty and preserves argmin precision).
// GEMM staging uses GLOBAL_LOAD_ASYNC_TO_LDS_B128 with double-buffered LDS so
// HBM/L2 streaming overlaps the WMMA pipeline (tracked via ASYNCcnt).
// ---------------------------------------------------------------------------

typedef __attribute__((ext_vector_type(2))) float v2f;
typedef __attribute__((ext_vector_type(8))) float v8f;

#ifndef RAVQ_N
#define RAVQ_N 16384
#define RAVQ_D 1024
#define RAVQ_K 8192
#endif

// Tiling: 256 threads = 8 waves; each wave owns 16 rows (BM=128); BN=64
// codes/cols per n-iteration (4 WMMA n-tiles/wave); KC=32 reduction chunk.
#define BM 128
#define BN 64
#define KC 32
#define PITCH (KC + 2)          // LDS pad: row stride 34 -> bank stride 2, no conflicts
#define NCH (RAVQ_D / KC)       // 32 reduction chunks
#define NPER 6                  // async b128 instrs / thread / chunk (4 X + 2 C)
#define BUF_STRIDE ((BM + BN) * PITCH)
#define SMEM2_BYTES (2 * BUF_STRIDE * 4)

// Buffer pointer helpers (pure offset arithmetic -- no LDS pointer aggregates,
// which ld.lld cannot statically initialize).
__device__ __forceinline__ float* xbuf(float* smem, int b) { return smem + b * BUF_STRIDE; }
__device__ __forceinline__ float* cbuf(float* smem, int b) { return smem + b * BUF_STRIDE + BM * PITCH; }

// ---------------------------------------------------------------------------
// Async global->LDS staging of a nrows x KC tile into a pitch-padded LDS
// buffer. One b128 = 4 contiguous floats inside one padded row. GVS
// addressing: 64-bit SGPR base + 32-bit per-lane byte offset.
// ---------------------------------------------------------------------------
__device__ __forceinline__ void async_stage_tile(float* ldsbuf, const float* gbase,
                                                 unsigned row0, unsigned k0, int nrows) {
    const unsigned lds_base = (unsigned)(uintptr_t)ldsbuf;
    const int nchunks = nrows * (KC / 4);
    #pragma unroll
    for (int i = threadIdx.x; i < nchunks; i += 256) {
        unsigned rr = (unsigned)i / (KC / 4);
        unsigned cc = ((unsigned)i % (KC / 4)) * 4u;
        unsigned lds_addr = lds_base + (rr * PITCH + cc) * 4u;
        unsigned goff = ((row0 + rr) * (unsigned)RAVQ_D + k0 + cc) * 4u;
        asm volatile("global_load_async_to_lds_b128 %0, %1, %2 offset:0"
                     :: "v"(lds_addr), "v"(goff), "s"(gbase)
                     : "memory");
    }
}

template <int N>
__device__ __forceinline__ void wait_asynccnt() {
    asm volatile("s_wait_asynccnt %0" :: "n"(N) : "memory");
}

// ---------------------------------------------------------------------------
// Kernel 0: zero counts (2*K ints) + 4 float sums
// ---------------------------------------------------------------------------
__global__ void init_ws_kernel(int* counts2, float* sums) {
    int i = blockIdx.x * blockDim.x + threadIdx.x;
    if (i < 2 * RAVQ_K) counts2[i] = 0;
    if (i < 4) sums[i] = 0.0f;
}

// ---------------------------------------------------------------------------
// Shared WMMA inner step: 8 k-steps x 4 n-tiles from one LDS buffer pair.
// A frag (16x4 f32): lane r=lane&15 holds row r, k = ks + 2*(lane>>4) + {0,1}
// B frag (4x16):     lane r holds col r, same k selector (NT GEMM, no transpose)
// ---------------------------------------------------------------------------
__device__ __forceinline__ void wmma_chunk(const float* Xs, const float* Cs,
                                           int row16, int r, int hi, v8f acc[4]) {
    #pragma unroll
    for (int ks = 0; ks < KC; ks += 4) {
        v2f a = *(const v2f*)&Xs[(row16 + r) * PITCH + ks + 2 * hi];
        #pragma unroll
        for (int t = 0; t < 4; ++t) {
            v2f b = *(const v2f*)&Cs[(t * 16 + r) * PITCH + ks + 2 * hi];
            acc[t] = __builtin_amdgcn_wmma_f32_16x16x4_f32(
                false, a, false, b, (short)0, acc[t], false, false);
        }
    }
}

// ---------------------------------------------------------------------------
// Kernel 1: Q[k, d] = sum_e emb[k, e] * W[d, e] + b[d]   (NT GEMM, WMMA f32)
// grid = (K/BM, D/BN), block = 256, double-buffered async LDS pipeline
// ---------------------------------------------------------------------------
__global__ __launch_bounds__(256) void codebook_gemm_kernel(
    const float* __restrict__ emb,   // [K, D]
    const float* __restrict__ W,     // [D, D]
    const float* __restrict__ bias,  // [D]
    float* __restrict__ Q)           // [K, D]
{
    extern __shared__ float smem[];

    const int tid   = threadIdx.x;
    const int lane  = tid & 31;
    const int wave  = tid >> 5;
    const unsigned m0 = blockIdx.x * BM;
    const unsigned n0 = blockIdx.y * BN;
    const int row16 = wave * 16;
    const int r     = lane & 15;
    const int hi    = lane >> 4;

    v8f acc[4] = {};

    // prologue: stage chunk 0 into buffer 0
    async_stage_tile(xbuf(smem, 0), emb, m0, 0, BM);
    async_stage_tile(cbuf(smem, 0), W,   n0, 0, BN);

    for (int ic = 0; ic < NCH; ++ic) {
        const int cur = ic & 1;
        if (ic + 1 < NCH) {
            async_stage_tile(xbuf(smem, cur ^ 1), emb, m0, (unsigned)(ic + 1) * KC, BM);
            async_stage_tile(cbuf(smem, cur ^ 1), W,   n0, (unsigned)(ic + 1) * KC, BN);
            wait_asynccnt<NPER>();   // chunk ic landed (in-order completion)
        } else {
            wait_asynccnt<0>();
        }
        __syncthreads();             // all waves have chunk ic resident
        wmma_chunk(xbuf(smem, cur), cbuf(smem, cur), row16, r, hi, acc);
        __syncthreads();             // all waves done reading before overwrite
    }

    // store: D-tile layout -> vgpr v: lanes0-15 row=v, lanes16-31 row=v+8; col=lane&15
    #pragma unroll
    for (int t = 0; t < 4; ++t) {
        int col = n0 + t * 16 + r;
        float bv = bias[col];
        #pragma unroll
        for (int v = 0; v < 8; ++v) {
            int rowk = m0 + row16 + v + 8 * hi;
            Q[(size_t)rowk * RAVQ_D + col] = acc[t][v] + bv;
        }
    }
}

// ---------------------------------------------------------------------------
// Kernel 2: cnorm[k] = sum_d Q[k,d]^2        grid = K blocks of 256
// ---------------------------------------------------------------------------
__global__ __launch_bounds__(256) void row_norm_kernel(
    const float* __restrict__ Q, float* __restrict__ cnorm)
{
    __shared__ float red[8];
    int k = blockIdx.x;
    float s = 0.0f;
    for (int d = threadIdx.x * 4; d < RAVQ_D; d += 256 * 4) {
        float4 q = *(const float4*)&Q[(size_t)k * RAVQ_D + d];
        s += q.x * q.x + q.y * q.y + q.z * q.z + q.w * q.w;
    }
    for (int off = 16; off >= 1; off >>= 1) s += __shfl_xor(s, off, 32);
    if ((threadIdx.x & 31) == 0) red[threadIdx.x >> 5] = s;
    __syncthreads();
    if (threadIdx.x == 0) {
        float t = 0.0f;
        #pragma unroll
        for (int w = 0; w < 8; ++w) t += red[w];
        cnorm[k] = t;
    }
}

// ---------------------------------------------------------------------------
// Kernel 3: fused distance-GEMM + argmin.
//   dist(i,c) = cnorm[c] - 2 * <x_i, q_c>   (||x||^2 dropped: row-constant)
// Running (min, idx) kept per lane-column slot in the WMMA C/D layout; one
// 16-wide butterfly per row at the end. grid = N/BM blocks of 256.
// ---------------------------------------------------------------------------
__global__ __launch_bounds__(256) void vq_argmin_kernel(
    const float* __restrict__ X,      // [N, D]
    const float* __restrict__ Q,      // [K, D]
    const float* __restrict__ cnorm,  // [K]
    int* __restrict__ idx_out,        // [N]
    int* __restrict__ counts)         // [K]
{
    extern __shared__ float smem[];

    const int tid   = threadIdx.x;
    const int lane  = tid & 31;
    const int wave  = tid >> 5;
    const unsigned m0 = blockIdx.x * BM;
    const int row16 = wave * 16;
    const int r     = lane & 15;
    const int hi    = lane >> 4;

    float runmin[8];
    int   runidx[8];
    #pragma unroll
    for (int v = 0; v < 8; ++v) { runmin[v] = 3.402823e38f; runidx[v] = 0; }

    for (unsigned n0 = 0; n0 < RAVQ_K; n0 += BN) {
        v8f acc[4] = {};

        async_stage_tile(xbuf(smem, 0), X, m0, 0, BM);
        async_stage_tile(cbuf(smem, 0), Q, n0, 0, BN);

        for (int ic = 0; ic < NCH; ++ic) {
            const int cur = ic & 1;
            if (ic + 1 < NCH) {
                async_stage_tile(xbuf(smem, cur ^ 1), X, m0, (unsigned)(ic + 1) * KC, BM);
                async_stage_tile(cbuf(smem, cur ^ 1), Q, n0, (unsigned)(ic + 1) * KC, BN);
                wait_asynccnt<NPER>();
            } else {
                wait_asynccnt<0>();
            }
            __syncthreads();
            wmma_chunk(xbuf(smem, cur), cbuf(smem, cur), row16, r, hi, acc);
            __syncthreads();
        }

        // running argmin update (ascending code order -> first-min semantics)
        #pragma unroll
        for (int t = 0; t < 4; ++t) {
            int code = n0 + t * 16 + r;
            float cn = cnorm[code];
            #pragma unroll
            for (int v = 0; v < 8; ++v) {
                float dist = cn - 2.0f * acc[t][v];
                if (dist < runmin[v]) { runmin[v] = dist; runidx[v] = code; }
            }
        }
    }

    // reduce over the 16 lane-columns inside each half-wave
    #pragma unroll
    for (int v = 0; v < 8; ++v) {
        float m = runmin[v];
        int   ix = runidx[v];
        #pragma unroll
        for (int off = 8; off >= 1; off >>= 1) {
            float om = __shfl_xor(m, off, 16);
            int   oi = __shfl_xor(ix, off, 16);
            if (om < m || (om == m && oi < ix)) { m = om; ix = oi; }
        }
        if (r == 0) {  // lanes 0 and 16: rows v and v+8 of this wave's strip
            int row = m0 + row16 + v + 8 * hi;
            idx_out[row] = ix;
            atomicAdd(&counts[ix], 1);
        }
    }
}

// ---------------------------------------------------------------------------
// Kernel 4: gather quantized rows (== straight-through outputs), accumulate
// the 4 cross-MSE sums. grid = N blocks of 256 (1 row / block, float4/thread).
// sums: [0]=sum(xs-qs)^2 [1]=sum(xr-qr)^2 [2]=sum(qr-xs)^2 [3]=sum(qs-xr)^2
// ---------------------------------------------------------------------------
__global__ __launch_bounds__(256) void gather_loss_kernel(
    const float* __restrict__ Xs_in,  // scRNA [N, D]
    const float* __restrict__ Xr_in,  // ribo  [N, D]
    const float* __restrict__ Q,      // [K, D]
    const int* __restrict__ idx_s,
    const int* __restrict__ idx_r,
    float* __restrict__ out_s,        // [N, D]
    float* __restrict__ out_r,        // [N, D]
    float* __restrict__ sums)         // [4]
{
    __shared__ float red[8][4];
    const int i   = blockIdx.x;
    const int d   = threadIdx.x * 4;
    const int is  = idx_s[i];
    const int ir  = idx_r[i];

    float4 qs = *(const float4*)&Q[(size_t)is * RAVQ_D + d];
    float4 qr = *(const float4*)&Q[(size_t)ir * RAVQ_D + d];
    float4 xs = *(const float4*)&Xs_in[(size_t)i * RAVQ_D + d];
    float4 xr = *(const float4*)&Xr_in[(size_t)i * RAVQ_D + d];

    *(float4*)&out_s[(size_t)i * RAVQ_D + d] = qs;
    *(float4*)&out_r[(size_t)i * RAVQ_D + d] = qr;

    float sA = 0, sB = 0, sC = 0, sD = 0;
    {
        float e;
        e = xs.x - qs.x; sA += e * e;  e = xs.y - qs.y; sA += e * e;
        e = xs.z - qs.z; sA += e * e;  e = xs.w - qs.w; sA += e * e;
        e = xr.x - qr.x; sB += e * e;  e = xr.y - qr.y; sB += e * e;
        e = xr.z - qr.z; sB += e * e;  e = xr.w - qr.w; sB += e * e;
        e = qr.x - xs.x; sC += e * e;  e = qr.y - xs.y; sC += e * e;
        e = qr.z - xs.z; sC += e * e;  e = qr.w - xs.w; sC += e * e;
        e = qs.x - xr.x; sD += e * e;  e = qs.y - xr.y; sD += e * e;
        e = qs.z - xr.z; sD += e * e;  e = qs.w - xr.w; sD += e * e;
    }
    for (int off = 16; off >= 1; off >>= 1) {
        sA += __shfl_xor(sA, off, 32); sB += __shfl_xor(sB, off, 32);
        sC += __shfl_xor(sC, off, 32); sD += __shfl_xor(sD, off, 32);
    }
    if ((threadIdx.x & 31) == 0) {
        int w = threadIdx.x >> 5;
        red[w][0] = sA; red[w][1] = sB; red[w][2] = sC; red[w][3] = sD;
    }
    __syncthreads();
    if (threadIdx.x == 0) {
        float a = 0, b = 0, c = 0, dd = 0;
        #pragma unroll
        for (int w = 0; w < 8; ++w) { a += red[w][0]; b += red[w][1]; c += red[w][2]; dd += red[w][3]; }
        atomicAdd(&sums[0], a); atomicAdd(&sums[1], b);
        atomicAdd(&sums[2], c); atomicAdd(&sums[3], dd);
    }
}

// ---------------------------------------------------------------------------
// Kernel 5: losses + perplexities. 1 block of 256.
// ---------------------------------------------------------------------------
__global__ __launch_bounds__(256) void finalize_kernel(
    const int* __restrict__ counts_s,
    const int* __restrict__ counts_r,
    const float* __restrict__ sums,
    float* __restrict__ out4)         // d_out + 2*N*D
{
    __shared__ float red[8][2];
    const float invN = 1.0f / (float)RAVQ_N;
    float es = 0.0f, er = 0.0f;
    for (int k = threadIdx.x; k < RAVQ_K; k += 256) {
        float ps = (float)counts_s[k] * invN;
        float pr = (float)counts_r[k] * invN;
        es -= ps * logf(ps + 1e-10f);
        er -= pr * logf(pr + 1e-10f);
    }
    for (int off = 16; off >= 1; off >>= 1) {
        es += __shfl_xor(es, off, 32); er += __shfl_xor(er, off, 32);
    }
    if ((threadIdx.x & 31) == 0) { red[threadIdx.x >> 5][0] = es; red[threadIdx.x >> 5][1] = er; }
    __syncthreads();
    if (threadIdx.x == 0) {
        float Es = 0, Er = 0;
        #pragma unroll
        for (int w = 0; w < 8; ++w) { Es += red[w][0]; Er += red[w][1]; }
        const float inv_nd = 1.0f / ((float)RAVQ_N * (float)RAVQ_D);
        float A = sums[0] * inv_nd;   // mse(xs, qs)
        float B = sums[1] * inv_nd;   // mse(xr, qr)
        float C = sums[2] * inv_nd;   // mse(qr, xs)
        float Dd = sums[3] * inv_nd;  // mse(qs, xr)
        float fwd = B + A + 0.5f * C + 0.5f * Dd;
        out4[0] = 0.5f * A;                 // 2*COMMIT*A, COMMIT=0.25
        out4[1] = 0.5f * B + 0.25f * fwd;   // 2*COMMIT*B + COMMIT*fwd
        out4[2] = expf(Es);
        out4[3] = expf(Er);
    }
}

// ---------------------------------------------------------------------------
// Host launcher
// ---------------------------------------------------------------------------
extern "C" void kernel_launch(void* const* d_in, const int* in_sizes, int n_in,
                              void* d_out, int out_size, void* d_ws, size_t ws_size,
                              hipStream_t stream) {
    (void)in_sizes; (void)n_in; (void)out_size; (void)ws_size;

    const float* scRNA = (const float*)d_in[0];   // [N, D]
    const float* ribo  = (const float*)d_in[1];   // [N, D]
    const float* emb   = (const float*)d_in[2];   // [K, D]
    const float* projw = (const float*)d_in[3];   // [D, D]
    const float* projb = (const float*)d_in[4];   // [D]
    // d_in[5] = flag (unused)

    // workspace layout (bytes)
    char* ws = (char*)d_ws;
    float* Q        = (float*)(ws);                                   // K*D f32 = 32 MB
    float* cnorm    = (float*)(ws + (size_t)RAVQ_K * RAVQ_D * 4);     // K
    int*   idx_s    = (int*)  ((char*)cnorm + (size_t)RAVQ_K * 4);    // N
    int*   idx_r    = (int*)  ((char*)idx_s + (size_t)RAVQ_N * 4);    // N
    int*   counts_s = (int*)  ((char*)idx_r + (size_t)RAVQ_N * 4);    // K
    int*   counts_r = (int*)  ((char*)counts_s + (size_t)RAVQ_K * 4); // K
    float* sums     = (float*)((char*)counts_r + (size_t)RAVQ_K * 4); // 4

    float* out_s = (float*)d_out;
    float* out_r = out_s + (size_t)RAVQ_N * RAVQ_D;
    float* out4  = out_r + (size_t)RAVQ_N * RAVQ_D;

    // 0) zero counts + sums
    init_ws_kernel<<<(2 * RAVQ_K + 255) / 256, 256, 0, stream>>>(counts_s, sums);

    // 1) codebook = emb @ W^T + b
    {
        dim3 grid(RAVQ_K / BM, RAVQ_D / BN);
        codebook_gemm_kernel<<<grid, 256, SMEM2_BYTES, stream>>>(emb, projw, projb, Q);
    }

    // 2) code norms
    row_norm_kernel<<<RAVQ_K, 256, 0, stream>>>(Q, cnorm);

    // 3) VQ search, both modalities
    vq_argmin_kernel<<<RAVQ_N / BM, 256, SMEM2_BYTES, stream>>>(scRNA, Q, cnorm, idx_s, counts_s);
    vq_argmin_kernel<<<RAVQ_N / BM, 256, SMEM2_BYTES, stream>>>(ribo,  Q, cnorm, idx_r, counts_r);

    // 4) gather quantized rows (straight-through outputs) + MSE sums
    gather_loss_kernel<<<RAVQ_N, 256, 0, stream>>>(scRNA, ribo, Q, idx_s, idx_r,
                                                   out_s, out_r, sums);

    // 5) losses + perplexities
    finalize_kernel<<<1, 256, 0, stream>>>(counts_s, counts_r, sums, out4);
}